// Aggregate_54571854463410
// MI455X (gfx1250) — compile-verified
//
#include <hip/hip_runtime.h>
#include <stdint.h>

typedef float v2f __attribute__((ext_vector_type(2)));
typedef float v4f __attribute__((ext_vector_type(4)));
typedef float v8f __attribute__((ext_vector_type(8)));

#define BZ 32
#define NNODES 8192
#define FDIM 256
#define ROWS_PER_WAVE 64
#define TILES_PER_WAVE (ROWS_PER_WAVE / 16)          // 4
#define WAVES_PER_BLOCK 4
#define THREADS1 (WAVES_PER_BLOCK * 32)              // 128
#define NWAVES ((BZ * NNODES) / ROWS_PER_WAVE)       // 4096
#define NBLOCKS1 (NWAVES / WAVES_PER_BLOCK)          // 1024
#define WPB (NNODES / ROWS_PER_WAVE)                 // 128 partials per graph
#define ROW_DW 260                                   // padded LDS row stride (dwords) -> bank-conflict-free
#define TILE_DW (16 * ROW_DW)                        // 4160 dwords = 16640 B per tile

// ---- async global -> LDS staging (CDNA5: GLOBAL_LOAD_ASYNC_TO_LDS_B128, ASYNCcnt) ----
__device__ __forceinline__ void async_copy_b128(uint32_t lds_byte_addr, const void* gaddr) {
    asm volatile("global_load_async_to_lds_b128 %0, %1, off"
                 :: "v"(lds_byte_addr), "v"(gaddr)
                 : "memory");
}
__device__ __forceinline__ void wait_async_le32() {
    asm volatile("s_wait_asynccnt 0x20" ::: "memory");
}
__device__ __forceinline__ void wait_async_zero() {
    asm volatile("s_wait_asynccnt 0x0" ::: "memory");
}

// Kernel 1: per-wave online-softmax partials over 64 contiguous rows.
// Gate dot products done with V_WMMA_F32_16X16X4_F32 (full f32 precision).
__global__ __launch_bounds__(THREADS1) void gate_partial_kernel(
    const float* __restrict__ x, const float* __restrict__ W1,
    float* __restrict__ H,      // [NWAVES][2]  (m, s)
    float* __restrict__ A)      // [NWAVES][FDIM] acc
{
    __shared__ float smem[WAVES_PER_BLOCK][2][TILE_DW]; // double-buffered tiles per wave
    __shared__ float w1s[FDIM];

    const int tid  = threadIdx.x;
    const int lane = tid & 31;
    const int wv   = tid >> 5;
    const int wid  = blockIdx.x * WAVES_PER_BLOCK + wv;

    for (int i = tid; i < FDIM; i += THREADS1) w1s[i] = W1[i];
    __syncthreads();

    const size_t row0  = (size_t)wid * ROWS_PER_WAVE;      // contiguous rows, one graph per wave
    const char*  gbase = (const char*)(x + row0 * FDIM);

    float* buf0 = &smem[wv][0][0];
    float* buf1 = &smem[wv][1][0];
    const uint32_t lds0 = (uint32_t)(uintptr_t)buf0;       // generic->LDS offset (low 32 bits)
    const uint32_t lds1 = (uint32_t)(uintptr_t)buf1;

    auto issue_tile = [&](uint32_t ldsbase, int t) {
        const char* g = gbase + (size_t)t * (16 * FDIM * 4);
#pragma unroll
        for (int j = 0; j < 32; ++j) {                     // 32 x (32 lanes x 16 B) = 16 KB tile
            uint32_t l = ldsbase + (uint32_t)(j * 512 + (j >> 1) * 16 + lane * 16);
            async_copy_b128(l, g + j * 512 + lane * 16);
        }
    };

    float m_run = -__builtin_inff();
    float s_run = 0.0f;
    float acc[8] = {0.f, 0.f, 0.f, 0.f, 0.f, 0.f, 0.f, 0.f};

    const int hf = lane >> 4;                              // half-wave id (K split for WMMA f32 A/B)

    issue_tile(lds0, 0);
    for (int t = 0; t < TILES_PER_WAVE; ++t) {
        const int cb = t & 1;
        const float* tile = cb ? buf1 : buf0;
        if (t + 1 < TILES_PER_WAVE) {
            issue_tile(cb ? lds0 : lds1, t + 1);           // prefetch next tile
            wait_async_le32();                             // current tile's 32 loads complete (in-order)
        } else {
            wait_async_zero();
        }

        // ---- gates: C[m,*] = sum_k x[row m,k] * W1[k] via 64 x WMMA f32 16x16x4 ----
        const float* arow = tile + (lane & 15) * ROW_DW + 2 * hf;  // A: lane m=row, K=(k0+2h, k0+2h+1)
        const float* brow = w1s + 2 * hf;                          // B: W1 chunk replicated over N
        v8f c = {0.f, 0.f, 0.f, 0.f, 0.f, 0.f, 0.f, 0.f};
#pragma unroll
        for (int k0 = 0; k0 < FDIM; k0 += 4) {
            v2f a = *(const v2f*)(arow + k0);
            v2f b = *(const v2f*)(brow + k0);
            c = __builtin_amdgcn_wmma_f32_16x16x4_f32(false, a, false, b, (short)0, c, false, false);
        }
        // lanes 0-15 hold gates of rows 0-7 in c[0..7]; lanes 16-31 hold rows 8-15.

        // ---- online softmax update ----
        float hmax = c[0];
#pragma unroll
        for (int i = 1; i < 8; ++i) hmax = fmaxf(hmax, c[i]);
        float tmax  = fmaxf(hmax, __shfl_xor(hmax, 16));
        float m_new = fmaxf(m_run, tmax);
        float resc  = __expf(m_run - m_new);               // exp(-inf)=0 on first tile
        s_run *= resc;
#pragma unroll
        for (int i = 0; i < 8; ++i) acc[i] *= resc;
        m_run = m_new;

        float w[8], hsum = 0.f;
#pragma unroll
        for (int i = 0; i < 8; ++i) { w[i] = __expf(c[i] - m_new); hsum += w[i]; }
        s_run += hsum + __shfl_xor(hsum, 16);

        float wrow[16];
#pragma unroll
        for (int i = 0; i < 8; ++i) {
            float wo = __shfl_xor(w[i], 16);               // other half's row weights
            wrow[i]     = hf ? wo   : w[i];
            wrow[8 + i] = hf ? w[i] : wo;
        }

        // ---- weighted accumulate: lane owns features 8*lane .. 8*lane+7 ----
        const float* frow = tile + lane * 8;
#pragma unroll
        for (int r = 0; r < 16; ++r) {
            v4f p0 = *(const v4f*)(frow + r * ROW_DW);
            v4f p1 = *(const v4f*)(frow + r * ROW_DW + 4);
            float wr = wrow[r];
#pragma unroll
            for (int i = 0; i < 4; ++i) { acc[i] += wr * p0[i]; acc[4 + i] += wr * p1[i]; }
        }
    }

    if (lane == 0) { H[2 * wid] = m_run; H[2 * wid + 1] = s_run; }
    float* ap = A + (size_t)wid * FDIM + lane * 8;
    v4f a0 = {acc[0], acc[1], acc[2], acc[3]};
    v4f a1 = {acc[4], acc[5], acc[6], acc[7]};
    *(v4f*)ap       = a0;
    *(v4f*)(ap + 4) = a1;
}

// Kernel 2: merge 128 per-wave partials per graph; normalize. Tiny (4 MB traffic total).
__global__ __launch_bounds__(FDIM) void combine_kernel(
    const float* __restrict__ H, const float* __restrict__ A, float* __restrict__ out)
{
    __shared__ float e[WPB];
    __shared__ float Ssh;
    const int b = blockIdx.x, tid = threadIdx.x;
    __shared__ float Msh;

    if (tid == 0) {
        float M = -__builtin_inff();
        for (int w = 0; w < WPB; ++w) M = fmaxf(M, H[2 * (b * WPB + w)]);
        float S = 0.f;
        for (int w = 0; w < WPB; ++w) S += H[2 * (b * WPB + w) + 1] * __expf(H[2 * (b * WPB + w)] - M);
        Msh = M; Ssh = S;
    }
    __syncthreads();
    if (tid < WPB) e[tid] = __expf(H[2 * (b * WPB + tid)] - Msh);
    __syncthreads();

    float sum = 0.f;
    const float* ab = A + (size_t)b * WPB * FDIM + tid;    // coalesced over tid
    for (int w = 0; w < WPB; ++w) sum += e[w] * ab[(size_t)w * FDIM];
    out[b * FDIM + tid] = sum / Ssh;
}

extern "C" void kernel_launch(void* const* d_in, const int* in_sizes, int n_in,
                              void* d_out, int out_size, void* d_ws, size_t ws_size,
                              hipStream_t stream) {
    const float* x  = (const float*)d_in[0];
    const float* W1 = (const float*)d_in[1];
    // d_in[2] = b1: softmax is shift-invariant -> mathematically irrelevant, not read.
    float* H = (float*)d_ws;                 // 2 * NWAVES floats
    float* A = (float*)d_ws + 2 * NWAVES;    // NWAVES * FDIM floats (16B aligned: 32 KB offset)
    float* out = (float*)d_out;

    hipLaunchKernelGGL(gate_partial_kernel, dim3(NBLOCKS1), dim3(THREADS1), 0, stream, x, W1, H, A);
    hipLaunchKernelGGL(combine_kernel, dim3(BZ), dim3(FDIM), 0, stream, H, A, out);
}